// FeatureMemoryBank_515396075780
// MI455X (gfx1250) — compile-verified
//
#include <hip/hip_runtime.h>
#include <hip/hip_bf16.h>

// ---------------- problem constants (from reference) ----------------
#define DIMK     256      // INPUT_SIZE
#define NMEM     100000   // OUTPUT_SIZE
#define NBATCH   1024     // BATCH
#define NNNUM    5
#define MOM      0.5f

// ---------------- tiling ----------------
#define RPB      16                        // batch rows per block (one WMMA M tile)
#define CPB      512                       // memory columns per block
#define NCHUNK   ((NMEM + CPB - 1) / CPB)  // 196 column chunks
#define NT       4                         // 16-col tiles per wave (8 waves * 64 = 512)
#define NSTAGE   4                         // operand pipeline depth (k-steps in flight)
#define NEGINF   (-3.0e38f)

typedef float v2f __attribute__((ext_vector_type(2)));
typedef float v8f __attribute__((ext_vector_type(8)));

// =====================================================================
// Pass 1: 16x512 sims tile via V_WMMA_F32_16X16X4_F32, fused local top-5
// =====================================================================
__global__ __launch_bounds__(256) void fmb_sims_topk_kernel(
    const float* __restrict__ x,      // [NBATCH, DIMK]
    const int*   __restrict__ y,      // [NBATCH]
    const float* __restrict__ mem,    // [NMEM, DIMK]  (pre-update memory)
    float* __restrict__ scrV,         // [NBATCH, NCHUNK, 5]
    int*   __restrict__ scrI)         // [NBATCH, NCHUNK, 5]
{
    __shared__ float sims[RPB][CPB];        // 32 KB
    __shared__ float candV[RPB][16][NNNUM]; // 5 KB
    __shared__ int   candI[RPB][16][NNNUM]; // 5 KB

    const int chunk   = blockIdx.x;
    const int rowTile = blockIdx.y;
    const int tid  = threadIdx.x;
    const int wave = tid >> 5;
    const int lane = tid & 31;
    const int l16  = lane & 15;
    const int hi   = lane >> 4;

    const int rowBase = rowTile * RPB;
    const int colBase = chunk * CPB;

    // ---------------- WMMA fp32 GEMM: sims[16][512] ----------------
    // A 16x4 f32 layout: lane holds row M=l16, k = hi*2 + {0,1} per VGPR
    // B 4x16 f32 layout: lane holds col N=l16, k = hi*2 + {0,1} per VGPR
    v8f zero = {0.f,0.f,0.f,0.f,0.f,0.f,0.f,0.f};
    v8f acc[NT];
    #pragma unroll
    for (int t = 0; t < NT; ++t) acc[t] = zero;

    const int kk0 = hi * 2;
    const float* xrow = x + (size_t)(rowBase + l16) * DIMK + kk0;
    const float* brow[NT];
    #pragma unroll
    for (int t = 0; t < NT; ++t) {
        int c  = colBase + wave * (NT * 16) + t * 16 + l16;
        int cc = (c < NMEM) ? c : (NMEM - 1);     // clamp (EXEC must stay all-1s)
        brow[t] = mem + (size_t)cc * DIMK + kk0;
    }

    // ---- modulo-scheduled K loop: NSTAGE operand sets live at once ----
    // Each unrolled sub-step CONSUMES stage s, then REFILLS stage s with
    // data for k + NSTAGE*4.  No register rotation copies exist, so the
    // compiler cannot collapse stages; each WMMA's operands were loaded
    // (NSTAGE-1)*5 loads earlier -> nonzero s_wait_loadcnt slack.
    v2f aS[NSTAGE];
    v2f bS[NSTAGE][NT];
    #pragma unroll
    for (int s = 0; s < NSTAGE; ++s) {
        aS[s] = *(const v2f*)(xrow + 4 * s);
        #pragma unroll
        for (int t = 0; t < NT; ++t) bS[s][t] = *(const v2f*)(brow[t] + 4 * s);
    }

    for (int k = 0; k < DIMK; k += 4 * NSTAGE) {
        #pragma unroll
        for (int s = 0; s < NSTAGE; ++s) {
            // consume stage s (k-step = k + 4*s)
            #pragma unroll
            for (int t = 0; t < NT; ++t) {
                // D = A(16x4) * B(4x16) + C   (exact fp32 matrix op)
                acc[t] = __builtin_amdgcn_wmma_f32_16x16x4_f32(
                    false, aS[s], false, bS[s][t], (short)0, acc[t], false, false);
            }
            // refill stage s with k-step = k + 4*NSTAGE + 4*s
            const int kr = k + 4 * NSTAGE + 4 * s;
            const int kn = (kr < DIMK) ? kr : 4 * s;   // tail: harmless dummy refill
            aS[s] = *(const v2f*)(xrow + kn);
            #pragma unroll
            for (int t = 0; t < NT; ++t) bS[s][t] = *(const v2f*)(brow[t] + kn);
        }
    }

    // C/D layout: VGPR r, lanes 0-15 -> M=r, N=l16 ; lanes 16-31 -> M=8+r
    #pragma unroll
    for (int t = 0; t < NT; ++t) {
        int lc  = wave * (NT * 16) + t * 16 + l16;
        bool oob = (colBase + lc) >= NMEM;
        #pragma unroll
        for (int r = 0; r < 8; ++r) {
            float v = acc[t][r];
            sims[r + 8 * hi][lc] = oob ? NEGINF : v;
        }
    }
    __syncthreads();

    // ---------------- fused per-chunk top-5 ----------------
    // 16 threads per row, 32 columns per thread
    {
        const int srow = tid >> 4;   // 0..15
        const int seg  = tid & 15;   // 0..15
        const int yv   = y[rowBase + srow];
        float bv[NNNUM]; int bi[NNNUM];
        #pragma unroll
        for (int i = 0; i < NNNUM; ++i) { bv[i] = NEGINF; bi[i] = -1; }
        for (int c = seg * 32; c < seg * 32 + 32; ++c) {
            int   gcol = colBase + c;
            float v    = sims[srow][c];
            if (gcol == yv) v = -2.0f;             // self-mask, as in reference
            if (v > bv[NNNUM - 1]) {
                int p = NNNUM - 1;
                while (p > 0 && v > bv[p - 1]) { bv[p] = bv[p-1]; bi[p] = bi[p-1]; --p; }
                bv[p] = v; bi[p] = gcol;
            }
        }
        #pragma unroll
        for (int i = 0; i < NNNUM; ++i) { candV[srow][seg][i] = bv[i]; candI[srow][seg][i] = bi[i]; }
    }
    __syncthreads();

    // merge 16 segment lists -> 5 candidates per (row, chunk)
    if (tid < RPB) {
        const int row = tid;
        float fv[NNNUM]; int fi[NNNUM];
        #pragma unroll
        for (int i = 0; i < NNNUM; ++i) { fv[i] = NEGINF; fi[i] = -1; }
        for (int s = 0; s < 16; ++s) {
            #pragma unroll
            for (int i = 0; i < NNNUM; ++i) {
                float v = candV[row][s][i]; int ix = candI[row][s][i];
                if (v > fv[NNNUM - 1]) {
                    int p = NNNUM - 1;
                    while (p > 0 && v > fv[p - 1]) { fv[p] = fv[p-1]; fi[p] = fi[p-1]; --p; }
                    fv[p] = v; fi[p] = ix;
                }
            }
        }
        size_t base = ((size_t)(rowBase + row) * NCHUNK + chunk) * NNNUM;
        #pragma unroll
        for (int i = 0; i < NNNUM; ++i) { scrV[base + i] = fv[i]; scrI[base + i] = fi[i]; }
    }
}

// =====================================================================
// Pass 2: merge per-chunk candidates -> final top-5; emit [y, idx0..4]
// =====================================================================
__global__ __launch_bounds__(256) void fmb_merge_kernel(
    const float* __restrict__ scrV,
    const int*   __restrict__ scrI,
    const int*   __restrict__ y,
    float*       __restrict__ nn_out)   // [NBATCH, 1+NNNUM] as float
{
    int row = blockIdx.x * blockDim.x + threadIdx.x;
    if (row >= NBATCH) return;
    float fv[NNNUM]; int fi[NNNUM];
    #pragma unroll
    for (int i = 0; i < NNNUM; ++i) { fv[i] = NEGINF; fi[i] = -1; }
    const float* v  = scrV + (size_t)row * NCHUNK * NNNUM;
    const int*   ix = scrI + (size_t)row * NCHUNK * NNNUM;
    for (int j = 0; j < NCHUNK * NNNUM; ++j) {
        float val = v[j]; int id = ix[j];
        if (val > fv[NNNUM - 1]) {
            int p = NNNUM - 1;
            while (p > 0 && val > fv[p - 1]) { fv[p] = fv[p-1]; fi[p] = fi[p-1]; --p; }
            fv[p] = val; fi[p] = id;
        }
    }
    nn_out[(size_t)row * (1 + NNNUM)] = (float)y[row];
    #pragma unroll
    for (int i = 0; i < NNNUM; ++i)
        nn_out[(size_t)row * (1 + NNNUM) + 1 + i] = (float)fi[i];
}

// =====================================================================
// Update: rows y[b] <- normalize(0.5*mem[y[b]] + 0.5*x[b]); 1 wave / row
// =====================================================================
__global__ __launch_bounds__(256) void fmb_update_kernel(
    const float* __restrict__ x,
    const int*   __restrict__ y,
    const float* __restrict__ mem,      // pre-update memory (input)
    float*       __restrict__ out_mem)  // copy of memory in d_out
{
    const int lane = threadIdx.x & 31;
    const int w    = threadIdx.x >> 5;
    const int b    = blockIdx.x * 8 + w;
    if (b >= NBATCH) return;
    const size_t row = (size_t)y[b];

    float v[8]; float ss = 0.f;
    #pragma unroll
    for (int j = 0; j < 8; ++j) {
        int c = j * 32 + lane;                    // coalesced 128B per step
        float wv = MOM * mem[row * DIMK + c] + (1.0f - MOM) * x[(size_t)b * DIMK + c];
        v[j] = wv; ss += wv * wv;
    }
    #pragma unroll
    for (int off = 16; off > 0; off >>= 1)        // wave32 butterfly reduce
        ss += __shfl_xor(ss, off, 32);
    float sn = sqrtf(ss);
    #pragma unroll
    for (int j = 0; j < 8; ++j) {
        int c = j * 32 + lane;
        out_mem[row * DIMK + c] = v[j] / sn;
    }
}

// =====================================================================
extern "C" void kernel_launch(void* const* d_in, const int* in_sizes, int n_in,
                              void* d_out, int out_size, void* d_ws, size_t ws_size,
                              hipStream_t stream) {
    const float* x   = (const float*)d_in[0];   // [1024, 256]
    const int*   y   = (const int*)  d_in[1];   // [1024]
    const float* mem = (const float*)d_in[2];   // [100000, 256]
    (void)in_sizes; (void)n_in; (void)out_size; // epoch = d_in[3] is statically 21 (kNN path)

    float* nn_out  = (float*)d_out;                         // 6144 floats
    float* mem_out = (float*)d_out + (size_t)NBATCH * (1 + NNNUM);

    // scratch for per-chunk top-5 candidates (values + indices)
    const size_t scrElems = (size_t)NBATCH * NCHUNK * NNNUM; // 1,003,520
    float* scrV; int* scrI;
    if (ws_size >= scrElems * 8) {
        scrV = (float*)d_ws;
        scrI = (int*)((float*)d_ws + scrElems);
    } else {
        // carve from the d_out memory region; it is overwritten afterwards
        scrV = mem_out;
        scrI = (int*)(mem_out + scrElems);
    }

    // 1) GEMM (WMMA fp32) + fused per-chunk top-5 (reads PRE-update memory)
    dim3 g1(NCHUNK, NBATCH / RPB);
    fmb_sims_topk_kernel<<<g1, 256, 0, stream>>>(x, y, mem, scrV, scrI);

    // 2) merge candidates -> nn indices
    fmb_merge_kernel<<<(NBATCH + 255) / 256, 256, 0, stream>>>(scrV, scrI, y, nn_out);

    // 3) new_memory = memory (bulk D2D copy, overwrites scratch region)
    hipMemcpyAsync(mem_out, mem, (size_t)NMEM * DIMK * sizeof(float),
                   hipMemcpyDeviceToDevice, stream);

    // 4) EMA + L2-renormalize the 1024 touched rows
    fmb_update_kernel<<<NBATCH / 8, 256, 0, stream>>>(x, y, mem, mem_out);
}